// LuongAttention_12232066859446
// MI455X (gfx1250) — compile-verified
//
#include <hip/hip_runtime.h>
#include <math.h>

typedef __attribute__((ext_vector_type(2))) float v2f;
typedef __attribute__((ext_vector_type(8))) float v8f;
typedef __attribute__((ext_vector_type(4))) unsigned int v4u;
typedef __attribute__((ext_vector_type(8))) int v8i;
typedef __attribute__((ext_vector_type(4))) int v4i;

#define BB 64
#define PP 4096
#define EE 512
#define DD 512
#define NCH 8            // chunks per batch
#define PCH (PP / NCH)   // 512 positions per chunk (one workgroup)
#define ST 64            // positions per TDM subtile (64 x 512 f32 = 128KB LDS)
#define NT (PCH / ST)    // 8 subtiles per chunk

// ---------------------------------------------------------------------------
// Kernel 1: proj[b,e] = sum_d H[b,d] * W[e,d]   (WMMA f32 16x16x4)
// ---------------------------------------------------------------------------
__global__ __launch_bounds__(32) void proj_gemm_wmma(const float* __restrict__ H,
                                                     const float* __restrict__ W,
                                                     float* __restrict__ proj) {
  const int m0   = blockIdx.x * 16;
  const int n0   = blockIdx.y * 16;
  const int lane = threadIdx.x;
  const int half = lane >> 4;
  const int id   = lane & 15;

  const float* ha = H + (size_t)(m0 + id) * DD + 2 * half;
  const float* wb = W + (size_t)(n0 + id) * DD + 2 * half;

  v8f c = {};
#pragma unroll 8
  for (int k0 = 0; k0 < DD; k0 += 4) {
    v2f a, b;
    a.x = ha[k0]; a.y = ha[k0 + 1];
    b.x = wb[k0]; b.y = wb[k0 + 1];
    c = __builtin_amdgcn_wmma_f32_16x16x4_f32(false, a, false, b, (short)0, c,
                                              false, false);
  }
#pragma unroll
  for (int i = 0; i < 8; ++i)
    proj[(size_t)(m0 + i + 8 * half) * EE + n0 + id] = c[i];
}

// ---------------------------------------------------------------------------
// TDM: async-load one ST x EE f32 tile (row-major, stride EE) into LDS.
// D# per CDNA5 ISA ch.8: group0 = count/lds_addr/global_addr/type,
// group1 = data_size + dims/strides, groups 2/3 zero (2-D tile).
// This toolchain uses the 6-arg builtin: (v4u, v8i, v4i, v4i, v8i, i32 cpol).
// ---------------------------------------------------------------------------
__device__ __forceinline__ void tdm_load_tile(const float* gsrc, float* ldst) {
  unsigned lds_addr = (unsigned)(uintptr_t)ldst;   // flat low 32 bits = LDS byte addr
  unsigned long long ga = (unsigned long long)(uintptr_t)gsrc;
  v4u g0 = { 1u,                                   // [1:0] count = 1 (valid)
             lds_addr,                             // [63:32] lds_addr
             (unsigned)(ga & 0xffffffffu),         // [95:64] global_addr lo
             (unsigned)((ga >> 32) & 0x01ffffffu)  // [120:96] global_addr hi
                 | (2u << 30) };                   // [127:126] type = 2 (image)
  v8i g1 = { (int)0x00020000,                      // data_size = 2 -> 4 bytes
             (int)(512u << 16),                    // tensor_dim0 = 512   [79:48]
             (int)((unsigned)ST << 16),            // tensor_dim1 = 64    [111:80]
             (int)(512u << 16),                    // tile_dim0 = 512     [127:112]
             (int)ST,                              // tile_dim1 = 64      [143:128]
             (int)512,                             // tensor_dim0_stride  [207:160]
             0, 0 };
  v4i z4 = { 0, 0, 0, 0 };
  v8i z8 = { 0, 0, 0, 0, 0, 0, 0, 0 };
  __builtin_amdgcn_tensor_load_to_lds(g0, g1, z4, z4, z8, 0);
}

// ---------------------------------------------------------------------------
// Kernel 2 (fused, flash-style): one workgroup = (batch b, chunk of 512 pos).
// TDM double-buffers 64x512 tiles into LDS; WMMA computes scores from LDS;
// online softmax (running max/sum) + context accumulation from the same tile.
// encoder_features is read from HBM exactly ONCE (~512MB -> ~22us @ 23.3TB/s).
// ---------------------------------------------------------------------------
__global__ __launch_bounds__(128) void fused_attn(const float* __restrict__ enc,
                                                  const float* __restrict__ proj,
                                                  float* __restrict__ raw_scores,
                                                  float* __restrict__ part,
                                                  float* __restrict__ chM,
                                                  float* __restrict__ chS) {
  extern __shared__ float smem[];
  float* buf0 = smem;                 // ST*EE floats
  float* buf1 = smem + ST * EE;       // ST*EE floats
  float* psh  = smem + 2 * ST * EE;   // EE floats: proj row
  float* swb  = psh + EE;             // ST floats: subtile scores/weights
  float* red  = swb + ST;             // 128 floats: reduction scratch

  const int tid  = threadIdx.x;
  const int ch   = blockIdx.x;
  const int b    = blockIdx.y;
  const int wv   = tid >> 5;          // wave id 0..3
  const int lane = tid & 31;
  const int half = lane >> 4;
  const int id   = lane & 15;

  // stage proj row into LDS (visible after the first barrier below)
  for (int i = tid; i < EE; i += 128) psh[i] = proj[(size_t)b * EE + i];

  const float* gbase = enc + ((size_t)b * PP + (size_t)ch * PCH) * EE;

  if (tid < 32) tdm_load_tile(gbase, buf0);   // prologue: tile 0 in flight

  float M = -__builtin_inff();
  float S = 0.f;
  float4 acc = make_float4(0.f, 0.f, 0.f, 0.f);

  for (int t = 0; t < NT; ++t) {
    float* bufc = (t & 1) ? buf1 : buf0;
    if (tid < 32) {
      if (t + 1 < NT) {
        // issue next tile into the buffer freed at the end of iteration t-1,
        // then wait until only that new load is still outstanding
        tdm_load_tile(gbase + (size_t)(t + 1) * ST * EE, (t & 1) ? buf0 : buf1);
        __builtin_amdgcn_s_wait_tensorcnt(1);
      } else {
        __builtin_amdgcn_s_wait_tensorcnt(0);   // drain last tile
      }
    }
    __syncthreads();   // tile t resident; psh ready (t==0)

    // ---- scores via WMMA from LDS: wave wv -> positions [wv*16, wv*16+16)
    {
      const float* arow = bufc + (size_t)(wv * 16 + id) * EE + 2 * half;
      const float* brow = psh + 2 * half;
      v8f c = {};
#pragma unroll 4
      for (int k0 = 0; k0 < EE; k0 += 4) {
        v2f a, bb;
        a.x  = arow[k0]; a.y  = arow[k0 + 1];
        bb.x = brow[k0]; bb.y = brow[k0 + 1];
        c = __builtin_amdgcn_wmma_f32_16x16x4_f32(false, a, false, bb, (short)0,
                                                  c, false, false);
      }
      if (id == 0) {   // divergence only after all WMMAs
#pragma unroll
        for (int i = 0; i < 8; ++i) {
          const int pt = wv * 16 + i + 8 * half;
          swb[pt] = c[i];
          raw_scores[(size_t)b * PP + (size_t)ch * PCH + (size_t)t * ST + pt] = c[i];
        }
      }
    }
    __syncthreads();

    // ---- online softmax update over this subtile's 64 scores
    const float sval = (tid < ST) ? swb[tid] : -__builtin_inff();
    red[tid] = sval;
    __syncthreads();
    for (int s = 64; s > 0; s >>= 1) {
      if (tid < s) red[tid] = fmaxf(red[tid], red[tid + s]);
      __syncthreads();
    }
    const float newM  = fmaxf(M, red[0]);
    const float scale = __expf(M - newM);
    acc.x *= scale; acc.y *= scale; acc.z *= scale; acc.w *= scale;
    const float wexp = (tid < ST) ? __expf(sval - newM) : 0.f;
    __syncthreads();   // done reading red[0]
    red[tid] = wexp;
    __syncthreads();
    for (int s = 64; s > 0; s >>= 1) {
      if (tid < s) red[tid] += red[tid + s];
      __syncthreads();
    }
    S = S * scale + red[0];
    M = newM;
    if (tid < ST) swb[tid] = wexp;
    __syncthreads();

    // ---- context accumulate: thread owns 4 consecutive e-columns
    const float4* ev = (const float4*)bufc;
#pragma unroll 4
    for (int p = 0; p < ST; ++p) {
      const float w  = swb[p];
      const float4 v = ev[p * (EE / 4) + tid];
      acc.x = fmaf(w, v.x, acc.x);
      acc.y = fmaf(w, v.y, acc.y);
      acc.z = fmaf(w, v.z, acc.z);
      acc.w = fmaf(w, v.w, acc.w);
    }
    __syncthreads();   // tile t free for reuse
  }

  float4* pp = (float4*)(part + ((size_t)b * NCH + ch) * EE);
  pp[tid] = acc;
  if (tid == 0) {
    chM[b * NCH + ch] = M;
    chS[b * NCH + ch] = S;
  }
}

// ---------------------------------------------------------------------------
// Kernel 3: merge the NCH chunk partials per batch -> context (B,E)
// ---------------------------------------------------------------------------
__global__ __launch_bounds__(512) void combine_ctx(const float* __restrict__ part,
                                                   const float* __restrict__ chM,
                                                   const float* __restrict__ chS,
                                                   float* __restrict__ ctx) {
  const int b = blockIdx.x;
  const int e = threadIdx.x;
  float Mg = -__builtin_inff();
#pragma unroll
  for (int c = 0; c < NCH; ++c) Mg = fmaxf(Mg, chM[b * NCH + c]);
  float Sg = 0.f, s = 0.f;
#pragma unroll
  for (int c = 0; c < NCH; ++c) {
    const float r = __expf(chM[b * NCH + c] - Mg);
    Sg += chS[b * NCH + c] * r;
    s  += part[((size_t)b * NCH + c) * EE + e] * r;
  }
  ctx[(size_t)b * EE + e] = s / Sg;
}

// ---------------------------------------------------------------------------
// Kernel 4: weights[b,p] = exp(score - Mg) / Sg  (in place over raw scores)
// ---------------------------------------------------------------------------
__global__ __launch_bounds__(256) void weights_final(float* __restrict__ wts,
                                                     const float* __restrict__ chM,
                                                     const float* __restrict__ chS) {
  const int b = blockIdx.y;
  const int p = blockIdx.x * 256 + threadIdx.x;
  float Mg = -__builtin_inff();
#pragma unroll
  for (int c = 0; c < NCH; ++c) Mg = fmaxf(Mg, chM[b * NCH + c]);
  float Sg = 0.f;
#pragma unroll
  for (int c = 0; c < NCH; ++c) Sg += chS[b * NCH + c] * __expf(chM[b * NCH + c] - Mg);
  const size_t idx = (size_t)b * PP + p;
  wts[idx] = __expf(wts[idx] - Mg) / Sg;
}

// ---------------------------------------------------------------------------
extern "C" void kernel_launch(void* const* d_in, const int* in_sizes, int n_in,
                              void* d_out, int out_size, void* d_ws, size_t ws_size,
                              hipStream_t stream) {
  const float* enc = (const float*)d_in[0];   // (B,P,E)
  const float* dh  = (const float*)d_in[1];   // (B,D)
  const float* W   = (const float*)d_in[2];   // (E,D)

  float* out = (float*)d_out;
  float* ctx = out;              // context: B*E floats (first tuple element)
  float* wts = out + BB * EE;    // weights: B*P floats (second tuple element)

  float* proj = (float*)d_ws;                 // B*E
  float* part = proj + BB * EE;               // B*NCH*E
  float* chM  = part + BB * NCH * EE;         // B*NCH
  float* chS  = chM + BB * NCH;               // B*NCH

  // 1) proj = H @ W^T  (WMMA)
  proj_gemm_wmma<<<dim3(BB / 16, EE / 16), 32, 0, stream>>>(dh, W, proj);

  // 2) fused flash-style pass: TDM -> LDS -> WMMA scores -> online softmax
  //    -> context partials.  Single 512MB HBM read.
  const size_t smem_bytes = (size_t)(2 * ST * EE + EE + ST + 128) * sizeof(float);
  fused_attn<<<dim3(NCH, BB), 128, smem_bytes, stream>>>(enc, proj, wts, part,
                                                         chM, chS);

  // 3) merge chunk partials -> context
  combine_ctx<<<BB, 512, 0, stream>>>(part, chM, chS, ctx);

  // 4) normalize stored raw scores -> weights
  weights_final<<<dim3(PP / 256, BB), 256, 0, stream>>>(wts, chM, chS);
}